// RadonTransform_1108101562468
// MI455X (gfx1250) — compile-verified
//
#include <hip/hip_runtime.h>

#ifndef __has_builtin
#define __has_builtin(x) 0
#endif

#define IMG_S   256
#define N_ANG   180
#define LSTRIDE 259                      // padded LDS row stride in floats (odd -> bank friendly)
#define LROWS   259                      // 1 top border row + 256 rows + 2 bottom border rows
#define LDS_FLOATS (LROWS * LSTRIDE)     // 67081
#define LDS_BYTES  (LDS_FLOATS * 4)      // 268324 bytes < 320 KB WGP LDS

typedef float        v2f __attribute__((ext_vector_type(2)));
typedef float        v8f __attribute__((ext_vector_type(8)));
typedef int          v4i __attribute__((ext_vector_type(4)));
typedef int          v8i __attribute__((ext_vector_type(8)));
typedef unsigned int v4u __attribute__((ext_vector_type(4)));

#if defined(__AMDGCN__) && __has_builtin(__builtin_amdgcn_tensor_load_to_lds) && __has_builtin(__builtin_amdgcn_s_wait_tensorcnt)
#define USE_TDM 1
#else
#define USE_TDM 0
#endif

#if defined(__AMDGCN__) && __has_builtin(__builtin_amdgcn_wmma_f32_16x16x4_f32)
#define USE_WMMA 1
#else
#define USE_WMMA 0
#endif

// Compile-time probes (warnings only; visible in stderr, do not fail the build)
#if defined(__AMDGCN__)
#if USE_TDM
#if __clang_major__ >= 23
#warning "CDNA5 probe: TDM path ENABLED (6-arg tensor_load_to_lds, clang>=23)"
#else
#warning "CDNA5 probe: TDM path ENABLED (5-arg tensor_load_to_lds, clang<23)"
#endif
#else
#warning "CDNA5 probe: TDM path DISABLED - fallback global->LDS copy in use"
#endif
#if !USE_WMMA
#warning "CDNA5 probe: WMMA f32_16x16x4 builtin NOT available - scalar reduction in use"
#endif
#endif

// Bilinear sample from the bordered LDS tile. px,py are pixel coords in the
// original 256x256 image; border cells are zero so clamping to [-1,256]
// exactly reproduces grid_sample zeros-padding (weights of clipped corners
// become 0, or the corner lands on a zero border cell).
__device__ __forceinline__ float bilinear_lds(const float* __restrict__ sm, float px, float py) {
    px = fminf(fmaxf(px, -1.0f), 256.0f);
    py = fminf(fmaxf(py, -1.0f), 256.0f);
    float x0 = floorf(px);
    float y0 = floorf(py);
    float wx = px - x0;
    float wy = py - y0;
    int ix = (int)x0 + 1;                 // +1: border offset
    int iy = (int)y0 + 1;
    const float* p = sm + iy * LSTRIDE + ix;
    float v00 = p[0];
    float v10 = p[1];
    float v01 = p[LSTRIDE];
    float v11 = p[LSTRIDE + 1];
    float top = v00 + wx * (v10 - v00);
    float bot = v01 + wx * (v11 - v01);
    return top + wy * (bot - top);
}

__global__ void __launch_bounds__(256) radon_fused(const float* __restrict__ img,
                                                   float* __restrict__ out) {
    extern __shared__ float sm[];
    const int a   = blockIdx.x;          // angle
    const int b   = blockIdx.y;          // batch
    const int tid = threadIdx.x;

    // ---- stage 1: zero ONLY the border cells (the interior is fully written
    // by the copy; out-of-circle interior pixels are zeroed by the mask pass).
    // Border = full rows 0,257,258 plus cols 0,257,258 of rows 1..256 — the
    // latter are exactly the TDM pad-skip cells.
    for (int i = tid; i < 3 * LSTRIDE; i += 256) {          // rows 0, 257, 258
        int rr = i / LSTRIDE, cc = i % LSTRIDE;
        int row = (rr == 0) ? 0 : (256 + rr);
        sm[row * LSTRIDE + cc] = 0.0f;
    }
    for (int i = tid; i < 3 * 256; i += 256) {              // cols 0,257,258 of rows 1..256
        int r  = i / 3 + 1;
        int c3 = i % 3;
        int col = (c3 == 0) ? 0 : (256 + c3);
        sm[r * LSTRIDE + col] = 0.0f;
    }
    __syncthreads();

    const float* __restrict__ src = img + (size_t)b * (IMG_S * IMG_S);

    // ---- stage 2: image -> LDS interior at stride 259 ----
#if USE_TDM
    if (tid < 32) {  // one wave drives the Tensor Data Mover
        unsigned long long ga = (unsigned long long)(const void*)src;
        unsigned int la = (unsigned int)(unsigned long long)(const void*)(sm + LSTRIDE + 1);
        v4u g0;
        g0.x = 1u;                                     // count=1 (valid user descriptor)
        g0.y = la;                                     // lds_addr (bytes)
        g0.z = (unsigned int)(ga & 0xffffffffull);     // global_addr[31:0]
        g0.w = (unsigned int)((ga >> 32) & 0x1ffffffull) | 0x80000000u; // addr[56:32] | type=2
        v8i g1;
        // data_size=4B (2<<16), pad_enable (1<<20), pad_interval=256dw (7<<22),
        // pad_amount=3dw (2<<25)  => LDS row stride becomes 256+3 = 259 dwords
        g1[0] = (int)((2u << 16) | (1u << 20) | (7u << 22) | (2u << 25));
        g1[1] = (int)(256u << 16);   // tensor_dim0[15:0] = 256
        g1[2] = (int)(256u << 16);   // tensor_dim1[15:0] = 256
        g1[3] = (int)(256u << 16);   // tile_dim0 = 256
        g1[4] = 256;                 // tile_dim1 = 256
        g1[5] = 256;                 // tensor_dim0_stride = 256
        g1[6] = 0;
        g1[7] = 0;
        v4i g2 = {0, 0, 0, 0};
        v4i g3 = {0, 0, 0, 0};
#if __clang_major__ >= 23
        v8i g4 = {0, 0, 0, 0, 0, 0, 0, 0};
        __builtin_amdgcn_tensor_load_to_lds(g0, g1, g2, g3, g4, 0);
#else
        __builtin_amdgcn_tensor_load_to_lds(g0, g1, g2, g3, 0);
#endif
        __builtin_amdgcn_s_wait_tensorcnt((short)0);
    }
    __syncthreads();
#else
    for (int i = tid; i < IMG_S * IMG_S; i += 256) {
        int r = i >> 8, c = i & 255;
        sm[(r + 1) * LSTRIDE + (c + 1)] = src[i];
    }
    __syncthreads();
#endif

    // ---- stage 3: inscribed-circle mask (exact integer test) ----
    for (int i = tid; i < IMG_S * IMG_S; i += 256) {
        int yy = i >> 8, xx = i & 255;
        int dx = xx - 128, dy = yy - 128;
        if (dx * dx + dy * dy > 16384)
            sm[(yy + 1) * LSTRIDE + (xx + 1)] = 0.0f;
    }
    __syncthreads();

    // ---- stage 4: ray sums.  px = 127.5(1-c+s) + c*i - s*j ; py = 127.5(1-c-s) + s*i + c*j
    const float ang = (float)a * (3.14159265358979323846f / 179.0f); // linspace(0,180,180) deg
    const float cs = cosf(ang);
    const float sn = sinf(ang);

    const int lane = tid & 31;
    const int wave = tid >> 5;
    const int m    = lane & 15;   // WMMA A-matrix row (j within the wave's 16-j group)
    const int kg   = lane >> 4;   // lanes 16-31 supply K=2,3 per the 16x4 f32 A layout

    float* __restrict__ orowBase = out + ((size_t)b * N_ANG + a) * IMG_S;

    for (int half = 0; half < 2; ++half) {
        const int   jbase = half * 128 + wave * 16;
        const float jf    = (float)(jbase + m);
        const float px0 = 127.5f * (1.0f - cs + sn) - sn * jf;
        const float py0 = 127.5f * (1.0f - cs - sn) + cs * jf;

#if USE_WMMA
        v8f acc = {0.f, 0.f, 0.f, 0.f, 0.f, 0.f, 0.f, 0.f};
#else
        float accs = 0.0f;
#endif
        for (int t = 0; t < 64; ++t) {
            const float i0 = (float)(4 * t + 2 * kg);
            const float i1 = i0 + 1.0f;
            float a0 = bilinear_lds(sm, px0 + cs * i0, py0 + sn * i0);
            float a1 = bilinear_lds(sm, px0 + cs * i1, py0 + sn * i1);
#if USE_WMMA
            // A[16x4] = samples (lane m = row j, K = 2*kg + {0,1}), B = ones(4x16)
            // => D[m,n] += sum_k A[m,k] : full-precision f32 row reduction on the matrix pipe.
            v2f Am = {a0, a1};
            v2f Bm = {1.0f, 1.0f};
            acc = __builtin_amdgcn_wmma_f32_16x16x4_f32(false, Am, false, Bm,
                                                        (short)0, acc, false, false);
#else
            accs += a0 + a1;
#endif
        }

        float* orow = orowBase + jbase;
#if USE_WMMA
        // C/D layout: vgpr r, lanes 0-15 -> M=r ; lanes 16-31 -> M=r+8.
        if (lane == 0) {
            orow[0] = acc[0]; orow[1] = acc[1]; orow[2] = acc[2]; orow[3] = acc[3];
            orow[4] = acc[4]; orow[5] = acc[5]; orow[6] = acc[6]; orow[7] = acc[7];
        } else if (lane == 16) {
            orow[8]  = acc[0]; orow[9]  = acc[1]; orow[10] = acc[2]; orow[11] = acc[3];
            orow[12] = acc[4]; orow[13] = acc[5]; orow[14] = acc[6]; orow[15] = acc[7];
        }
#else
        accs += __shfl_xor(accs, 16);
        if (lane < 16) orow[m] = accs;
#endif
    }
}

extern "C" void kernel_launch(void* const* d_in, const int* in_sizes, int n_in,
                              void* d_out, int out_size, void* d_ws, size_t ws_size,
                              hipStream_t stream) {
    (void)n_in; (void)out_size; (void)d_ws; (void)ws_size;
    const float* x = (const float*)d_in[0];
    float* out = (float*)d_out;
    int B = in_sizes[0] / (IMG_S * IMG_S);
    if (B <= 0) B = 1;
    (void)hipFuncSetAttribute((const void*)radon_fused,
                              hipFuncAttributeMaxDynamicSharedMemorySize,
                              (int)LDS_BYTES);
    dim3 grid(N_ANG, B);
    radon_fused<<<grid, dim3(256), (size_t)LDS_BYTES, stream>>>(x, out);
}